// Entangle_81475529605497
// MI455X (gfx1250) — compile-verified
//
#include <hip/hip_runtime.h>
#include <hip/hip_bf16.h>

#define DB 4
#define DN 8
#define DC 11
#define DS 256
#define BN 32   /* DB*DN rows per GEMM */

typedef float v2f __attribute__((ext_vector_type(2)));
typedef float v8f __attribute__((ext_vector_type(8)));

// V_WMMA_F32_16X16X4_F32: D = A(16x4) x B(4x16) + C(16x16), wave32.
static __device__ __forceinline__ v8f wmma4(v2f a, v2f b, v8f c) {
  return __builtin_amdgcn_wmma_f32_16x16x4_f32(false, a, false, b, (short)0, c, false, false);
}

// ---------------- K1: DFT twiddle tables c[t][k]=cos(2*pi*t*k/S), s=sin ----
__global__ __launch_bounds__(256) void tables_kernel(float* __restrict__ ct,
                                                     float* __restrict__ st) {
  int t = blockIdx.x, k = threadIdx.x;
  float a = (float)(t * k) * (1.0f / 128.0f);  // 2*t*k/256, exact in fp32
  ct[t * DS + k] = cospif(a);
  st[t * DS + k] = sinpif(a);
}

// ---------------- K0: zero y (we accumulate with atomics) ------------------
__global__ __launch_bounds__(256) void zero_y_kernel(float* __restrict__ y) {
  int i = blockIdx.x * 256 + threadIdx.x;
  if (i < DB * DN * DC * DS) y[i] = 0.0f;
}

// ---------------- K2: sig = fft(x) as DFT matmul (x real) ------------------
__global__ __launch_bounds__(256) void dft_x_kernel(const float* __restrict__ x,
                                                    const float* __restrict__ ct,
                                                    const float* __restrict__ st,
                                                    float* __restrict__ sr,
                                                    float* __restrict__ si) {
  __shared__ float xs[DS];
  int row = blockIdx.x;           // (b*N+i)*C + c
  int k = threadIdx.x;
  xs[k] = x[row * DS + k];
  __syncthreads();
  float ar = 0.0f, ai = 0.0f;
  for (int t = 0; t < DS; ++t) {
    float xv = xs[t];
    ar = fmaf(xv, ct[t * DS + k], ar);
    ai = fmaf(-xv, st[t * DS + k], ai);   // fft uses e^{-2pi i kt/S}
  }
  sr[row * DS + k] = ar;
  si[row * DS + k] = ai;
}

// ---------------- K3: mask = softmax_s(|k|) * (kr + i ki)/|k| --------------
__global__ __launch_bounds__(256) void mask_kernel(const float* __restrict__ kr,
                                                   const float* __restrict__ ki,
                                                   float* __restrict__ mr,
                                                   float* __restrict__ mi2) {
  size_t base = (size_t)blockIdx.x * DS * DS;  // block = j*C + c
  int t = threadIdx.x;
  float mx = -3.4e38f;
  for (int s = 0; s < DS; ++s) {
    float a = kr[base + (size_t)s * DS + t];
    float b = ki[base + (size_t)s * DS + t];
    mx = fmaxf(mx, sqrtf(a * a + b * b));
  }
  float sum = 0.0f;
  for (int s = 0; s < DS; ++s) {
    float a = kr[base + (size_t)s * DS + t];
    float b = ki[base + (size_t)s * DS + t];
    sum += expf(sqrtf(a * a + b * b) - mx);
  }
  float inv = 1.0f / sum;
  for (int s = 0; s < DS; ++s) {
    float a = kr[base + (size_t)s * DS + t];
    float b = ki[base + (size_t)s * DS + t];
    float mg = sqrtf(a * a + b * b);
    float wgt = expf(mg - mx) * inv;
    float co = (mg > 0.0f) ? a / mg : 1.0f;  // cos(atan2(0,0)) = 1
    float sn = (mg > 0.0f) ? b / mg : 0.0f;
    mr[base + (size_t)s * DS + t] = wgt * co;
    mi2[base + (size_t)s * DS + t] = wgt * sn;
  }
}

// ---------------- K4: Bt[b,j,c,s] = sum_t sig[b,j,c,t]*mask[j,c,s,t] -------
__global__ __launch_bounds__(256) void bt_kernel(const float* __restrict__ sr,
                                                 const float* __restrict__ si,
                                                 const float* __restrict__ mr,
                                                 const float* __restrict__ mi2,
                                                 float* __restrict__ btr,
                                                 float* __restrict__ bti) {
  int jj = blockIdx.x / DC, c = blockIdx.x % DC;
  int lane = threadIdx.x & 31, wv = threadIdx.x >> 5;
  size_t mbase = (size_t)(jj * DC + c) * DS * DS;
  for (int s = wv; s < DS; s += 8) {
    float ar[DB] = {0, 0, 0, 0}, ai[DB] = {0, 0, 0, 0};
    for (int t = lane; t < DS; t += 32) {
      float m_r = mr[mbase + (size_t)s * DS + t];
      float m_i = mi2[mbase + (size_t)s * DS + t];
#pragma unroll
      for (int b = 0; b < DB; ++b) {
        int row = (b * DN + jj) * DC + c;
        float zr = sr[row * DS + t], zi = si[row * DS + t];
        ar[b] += zr * m_r - zi * m_i;
        ai[b] += zr * m_i + zi * m_r;
      }
    }
#pragma unroll
    for (int off = 16; off > 0; off >>= 1) {
#pragma unroll
      for (int b = 0; b < DB; ++b) {
        ar[b] += __shfl_xor(ar[b], off, 32);
        ai[b] += __shfl_xor(ai[b], off, 32);
      }
    }
    if (lane == 0) {
#pragma unroll
      for (int b = 0; b < DB; ++b) {
        btr[((b * DN + jj) * DC + c) * DS + s] = ar[b];
        bti[((b * DN + jj) * DC + c) * DS + s] = ai[b];
      }
    }
  }
}

// ---------------- K5: M + s_out (184 MB streaming write) -------------------
__global__ __launch_bounds__(256) void sout_kernel(const float* __restrict__ sr,
                                                   const float* __restrict__ si,
                                                   const float* __restrict__ mr,
                                                   const float* __restrict__ mi2,
                                                   float* __restrict__ out) {
  int c = blockIdx.x / DS, s = blockIdx.x % DS;
  int t = threadIdx.x;
  float Mr[DB] = {0, 0, 0, 0}, Mi[DB] = {0, 0, 0, 0};
  for (int j = 0; j < DN; ++j) {
    size_t midx = (size_t)(j * DC + c) * DS * DS + (size_t)s * DS + t;
    float m_r = mr[midx], m_i = mi2[midx];
#pragma unroll
    for (int b = 0; b < DB; ++b) {
      int row = (b * DN + j) * DC + c;
      float zr = sr[row * DS + t], zi = si[row * DS + t];
      Mr[b] += zr * m_r - zi * m_i;
      Mi[b] += zr * m_i + zi * m_r;
    }
  }
#pragma unroll
  for (int b = 0; b < DB; ++b)
    for (int j = 0; j < DN; ++j) {
      int row = (b * DN + j) * DC + c;
      float zr = sr[row * DS + s], zi = si[row * DS + s];  // scalar broadcast
      float o_r = (zr * Mr[b] - zi * Mi[b]) * 0.125f;      // /N
      float o_i = (zr * Mi[b] + zi * Mr[b]) * 0.125f;
      size_t idx = ((size_t)row * DS + s) * DS + t;
      float2 v; v.x = o_r; v.y = o_i;
      ((float2*)out)[idx] = v;  // interleaved complex64
    }
}

// ---------------- K6: mix via exact DFT identity ---------------------------
// corr[b,i,j,c] = S * x[b,i,c,0] * x[b,j,c,0]  (mean of ifft == prod[...,0]/S)
__global__ __launch_bounds__(256) void mix_kernel(const float* __restrict__ x,
                                                  const float* __restrict__ gm,
                                                  const float* __restrict__ gl,
                                                  const float* __restrict__ gh,
                                                  float* __restrict__ mixw) {
  int idx = blockIdx.x * 256 + threadIdx.x;
  if (idx >= DB * DN * DN * DC) return;
  int c = idx % DC; int r = idx / DC;
  int j = r % DN; r /= DN;
  int i = r % DN; int b = r / DN;
  float xi0 = x[((b * DN + i) * DC + c) * DS];
  float xj0 = x[((b * DN + j) * DC + c) * DS];
  float cm = (float)DS * fabsf(xi0 * xj0);
  float mean = gm[i];
  float sd = (cm < mean) ? expf(gl[i]) : expf(gh[i]);
  float d = (cm - mean) / sd;
  mixw[idx] = expf(-0.5f * d * d);
}

// ---------------- WMMA iDFT: E = col(32x256) x (c + i s)(256x256) ----------
static __device__ __forceinline__ void idft_gemm(const float* __restrict__ cRl,
                                                 const float* __restrict__ cIl,
                                                 const float* __restrict__ ct,
                                                 const float* __restrict__ st,
                                                 int kh, int lm, int tbase,
                                                 v8f eR[2][2], v8f eI[2][2]) {
  for (int k0 = 0; k0 < DS; k0 += 4) {
    v2f zr[2], zi[2];
#pragma unroll
    for (int mt = 0; mt < 2; ++mt) {
      int rrow = mt * 16 + lm;
      zr[mt] = (v2f){cRl[rrow * DS + k0 + kh], cRl[rrow * DS + k0 + kh + 1]};
      zi[mt] = (v2f){cIl[rrow * DS + k0 + kh], cIl[rrow * DS + k0 + kh + 1]};
    }
#pragma unroll
    for (int nt = 0; nt < 2; ++nt) {
      int u = tbase + nt * 16 + lm;
      v2f bc = (v2f){ct[(k0 + kh) * DS + u], ct[(k0 + kh + 1) * DS + u]};
      v2f bs = (v2f){st[(k0 + kh) * DS + u], st[(k0 + kh + 1) * DS + u]};
      v2f bsn = (v2f){-bs.x, -bs.y};
#pragma unroll
      for (int mt = 0; mt < 2; ++mt) {
        eR[mt][nt] = wmma4(zr[mt], bc, eR[mt][nt]);   // re += zr*c
        eR[mt][nt] = wmma4(zi[mt], bsn, eR[mt][nt]);  // re -= zi*s
        eI[mt][nt] = wmma4(zr[mt], bs, eI[mt][nt]);   // im += zr*s
        eI[mt][nt] = wmma4(zi[mt], bc, eI[mt][nt]);   // im += zi*c
      }
    }
  }
}

// ---------------- K7: fused A-GEMM -> col0 -> iDFT -> col1 -> iDFT -> y ----
__global__ __launch_bounds__(256) void fused_y_kernel(
    const float* __restrict__ sr, const float* __restrict__ si,
    const float* __restrict__ mr, const float* __restrict__ mi2,
    const float* __restrict__ btr, const float* __restrict__ bti,
    const float* __restrict__ mixw, const float* __restrict__ ct,
    const float* __restrict__ st, const float* __restrict__ x,
    const float* __restrict__ pol, float* __restrict__ y) {
  extern __shared__ float lds[];
  float* sR = lds;                 // [32][256] sig real
  float* sI = sR + BN * DS;        // [32][256] sig imag
  float* cR = sI + BN * DS;        // [32][256] col buffer real
  float* cI = cR + BN * DS;        // [32][256] col buffer imag
  __shared__ float cpol[DN], spol[DN], mixsh[BN];

  const int jj = blockIdx.x / DC;
  const int c = blockIdx.x % DC;
  const int tid = threadIdx.x;
  const int lane = tid & 31;
  const int wv = tid >> 5;         // 8 waves; wave owns 32 output columns
  const int lm = lane & 15;
  const int lh = lane >> 4;
  const int kh = lh * 2;
  const int tbase = wv * 32;

  for (int row = 0; row < BN; ++row) {
    sR[row * DS + tid] = sr[(row * DC + c) * DS + tid];
    sI[row * DS + tid] = si[(row * DC + c) * DS + tid];
  }
  if (tid < DN) { cpol[tid] = cosf(pol[tid]); spol[tid] = sinf(pol[tid]); }
  if (tid < BN) mixsh[tid] = mixw[(tid * DN + jj) * DC + c];
  __syncthreads();

  const float* mrb = mr + (size_t)(jj * DC + c) * DS * DS;
  const float* mib = mi2 + (size_t)(jj * DC + c) * DS * DS;

  // GEMM1: A[(b,i)][t] = sum_s sig[(b,i)][s] * mask[s][t]  (complex)
  v8f aR[2][2] = {}; v8f aI[2][2] = {};
  for (int k0 = 0; k0 < DS; k0 += 4) {
    v2f fr[2], fi[2];
#pragma unroll
    for (int mt = 0; mt < 2; ++mt) {
      int rrow = mt * 16 + lm;
      fr[mt] = (v2f){sR[rrow * DS + k0 + kh], sR[rrow * DS + k0 + kh + 1]};
      fi[mt] = (v2f){sI[rrow * DS + k0 + kh], sI[rrow * DS + k0 + kh + 1]};
    }
#pragma unroll
    for (int nt = 0; nt < 2; ++nt) {
      int t = tbase + nt * 16 + lm;
      v2f br = (v2f){mrb[(k0 + kh) * DS + t], mrb[(k0 + kh + 1) * DS + t]};
      v2f bi = (v2f){mib[(k0 + kh) * DS + t], mib[(k0 + kh + 1) * DS + t]};
      v2f bin = (v2f){-bi.x, -bi.y};
#pragma unroll
      for (int mt = 0; mt < 2; ++mt) {
        aR[mt][nt] = wmma4(fr[mt], br, aR[mt][nt]);
        aR[mt][nt] = wmma4(fi[mt], bin, aR[mt][nt]);
        aI[mt][nt] = wmma4(fr[mt], bi, aI[mt][nt]);
        aI[mt][nt] = wmma4(fi[mt], br, aI[mt][nt]);
      }
    }
  }

  // col0[m][t] = A[m][t] * sig[b(m)*N+j][t]
#pragma unroll
  for (int mt = 0; mt < 2; ++mt)
#pragma unroll
    for (int nt = 0; nt < 2; ++nt) {
      int t = tbase + nt * 16 + lm;
#pragma unroll
      for (int r = 0; r < 8; ++r) {
        int m = mt * 16 + 8 * lh + r;            // C/D layout: M = r + 8*(lane/16)
        int zrow = (m >> 3) * DN + jj;
        float zr = sR[zrow * DS + t], zi = sI[zrow * DS + t];
        float ar = aR[mt][nt][r], ai = aI[mt][nt][r];
        cR[m * DS + t] = ar * zr - ai * zi;
        cI[m * DS + t] = ar * zi + ai * zr;
      }
    }
  __syncthreads();

  v8f eR[2][2] = {}, eI[2][2] = {};
  idft_gemm(cR, cI, ct, st, kh, lm, tbase, eR, eI);

  float sm0v[2][2][8];
#pragma unroll
  for (int mt = 0; mt < 2; ++mt)
#pragma unroll
    for (int nt = 0; nt < 2; ++nt)
#pragma unroll
      for (int r = 0; r < 8; ++r) {
        float re = eR[mt][nt][r], im = eI[mt][nt][r];
        sm0v[mt][nt][r] = sqrtf(re * re + im * im) * (1.0f / DS);
      }
  __syncthreads();  // all waves done reading col0

  // col1[m][s] = sig[m][s] * Bt[b(m)*N+j][s]
  for (int m = 0; m < BN; ++m) {
    int btrow = ((m >> 3) * DN + jj) * DC + c;
    float br_ = btr[btrow * DS + tid], bi_ = bti[btrow * DS + tid];
    float xr = sR[m * DS + tid], xi = sI[m * DS + tid];
    cR[m * DS + tid] = xr * br_ - xi * bi_;
    cI[m * DS + tid] = xr * bi_ + xi * br_;
  }
  __syncthreads();

  v8f gR[2][2] = {}, gI[2][2] = {};
  idft_gemm(cR, cI, ct, st, kh, lm, tbase, gR, gI);

#pragma unroll
  for (int mt = 0; mt < 2; ++mt)
#pragma unroll
    for (int nt = 0; nt < 2; ++nt) {
      int u = tbase + nt * 16 + lm;
#pragma unroll
      for (int r = 0; r < 8; ++r) {
        int m = mt * 16 + 8 * lh + r;
        int i = m & 7;
        float re = gR[mt][nt][r], im = gI[mt][nt][r];
        float sm1 = sqrtf(re * re + im * im) * (1.0f / DS);
        float smear = cpol[i] * sm0v[mt][nt][r] + spol[i] * sm1;
        float mv = mixsh[m];
        float xv = x[(m * DC + c) * DS + u];
        atomicAdd(&y[(m * DC + c) * DS + u],
                  0.125f * (mv * smear + (1.0f - mv) * xv));
      }
    }
}

// ---------------------------------------------------------------------------
extern "C" void kernel_launch(void* const* d_in, const int* in_sizes, int n_in,
                              void* d_out, int out_size, void* d_ws, size_t ws_size,
                              hipStream_t stream) {
  const float* x   = (const float*)d_in[0];
  const float* kr  = (const float*)d_in[1];
  const float* ki  = (const float*)d_in[2];
  const float* gm  = (const float*)d_in[3];
  const float* gl  = (const float*)d_in[4];
  const float* gh  = (const float*)d_in[5];
  const float* pol = (const float*)d_in[6];

  // Workspace layout (floats), total ~48.1 MB
  float* ws = (float*)d_ws;
  float* ct   = ws;                  // 65536
  float* st   = ct + 65536;          // 65536
  float* sr   = st + 65536;          // 90112
  float* si   = sr + 90112;          // 90112
  float* mr   = si + 90112;          // 5767168
  float* mi2  = mr + 5767168;        // 5767168
  float* btr  = mi2 + 5767168;       // 90112
  float* bti  = btr + 90112;         // 90112
  float* mixw = bti + 90112;         // 2816

  float* y = (float*)d_out;          // 90112 floats
  float* sout = y + 90112;           // interleaved complex64, 2*23068672 floats

  tables_kernel<<<DS, DS, 0, stream>>>(ct, st);
  zero_y_kernel<<<(DB * DN * DC * DS) / 256, 256, 0, stream>>>(y);
  dft_x_kernel<<<DB * DN * DC, DS, 0, stream>>>(x, ct, st, sr, si);
  mask_kernel<<<DN * DC, DS, 0, stream>>>(kr, ki, mr, mi2);
  bt_kernel<<<DN * DC, 256, 0, stream>>>(sr, si, mr, mi2, btr, bti);
  sout_kernel<<<DC * DS, DS, 0, stream>>>(sr, si, mr, mi2, sout);
  mix_kernel<<<(DB * DN * DN * DC + 255) / 256, 256, 0, stream>>>(x, gm, gl, gh, mixw);
  fused_y_kernel<<<DN * DC, 256, 4 * BN * DS * sizeof(float), stream>>>(
      sr, si, mr, mi2, btr, bti, mixw, ct, st, x, pol, y);
}